// ChamferLoss_48825188221598
// MI455X (gfx1250) — compile-verified
//
#include <hip/hip_runtime.h>
#include <hip/hip_bf16.h>
#include <stdint.h>

// ---------------------------------------------------------------------------
// Chamfer loss for B=32, K=2048, D=16 on gfx1250 (MI455X).
// Core: V_WMMA_F32_16X16X4_F32 computes 16x16 dist^2 tiles directly:
//   A = -2 * pred tile (16x4, K=3 padded), C = psq[i] + tsq[j]  =>  D = dist^2
// Row min/argmin kept wave-local; column min via L2 atomicMin on uint-punned
// nonnegative floats. Everything is L2-resident; kernel is epilogue-bound.
// ---------------------------------------------------------------------------

#define B_  32
#define K_  2048
#define D_  16
#define TILES_ (K_ / 16)   // 128 column tiles of 16

typedef float v2f __attribute__((ext_vector_type(2)));
typedef float v8f __attribute__((ext_vector_type(8)));

#define INF_BITS 0x7F800000u

__global__ __launch_bounds__(256)
void chamfer_init_kernel(unsigned int* __restrict__ t2p_bits,
                         float* __restrict__ accum, int n)
{
    int i = blockIdx.x * blockDim.x + threadIdx.x;
    if (i < n) t2p_bits[i] = INF_BITS;         // +inf as ordered uint
    if (i < 4) accum[i] = 0.0f;
}

// One wave32 per (batch, 16-row pred tile). grid: B*K/16 waves.
__global__ __launch_bounds__(256)
void chamfer_wmma_kernel(const float* __restrict__ pred,       // (B,K,3)
                         const float* __restrict__ tgt,        // (B,K,3)
                         const unsigned char* __restrict__ mask, // (B,K) bool bytes
                         float* __restrict__ p2t,              // (B,K) min dist^2 (masked)
                         int*   __restrict__ amin,             // (B,K) argmin col
                         unsigned int* __restrict__ t2p_bits)  // (B,K) col-min bits
{
    const int wave = blockIdx.x * (blockDim.x >> 5) + (threadIdx.x >> 5);
    const int lane = threadIdx.x & 31;
    const int b    = wave >> 7;        // / (K/16 = 128)
    const int rt   = wave & 127;
    const int i0   = rt * 16;
    const int half = lane >> 4;        // 0: lanes 0-15 (K=0,1), 1: lanes 16-31 (K=2,3)
    const int l16  = lane & 15;

    // ---- A matrix (16x4 f32, ISA layout), pre-scaled by -2; psq per row ----
    const float* pc = pred + ((size_t)b * K_ + i0) * 3;
    float a0, a1, psq_part;
    if (!half) {
        float x = pc[l16 * 3 + 0], y = pc[l16 * 3 + 1];
        psq_part = x * x + y * y;
        a0 = -2.0f * x; a1 = -2.0f * y;
    } else {
        float z = pc[l16 * 3 + 2];
        psq_part = z * z;
        a0 = -2.0f * z; a1 = 0.0f;     // K=3 pad
    }
    float psq_all = psq_part + __shfl_xor(psq_part, 16, 32);   // psq[row = l16]
    float p8[8];
#pragma unroll
    for (int r = 0; r < 8; ++r)
        p8[r] = __shfl(psq_all, r + half * 8, 32);             // psq[r + 8*half]

    v2f A; A.x = a0; A.y = a1;

    float rowmin[8]; int rowidx[8];
#pragma unroll
    for (int r = 0; r < 8; ++r) { rowmin[r] = __builtin_inff(); rowidx[r] = 0; }

    const float*         tc  = tgt  + (size_t)b * K_ * 3;
    const unsigned char* mk  = mask + (size_t)b * K_;
    unsigned int*        t2p = t2p_bits + (size_t)b * K_;

    for (int jt = 0; jt < TILES_; ++jt) {
        const int j0 = jt * 16;

        // B matrix (4x16 f32, ISA layout): lane<16 -> (x,y) of col l16; lane>=16 -> (z,0)
        float b0, b1, tsq_part;
        if (!half) {
            float x = tc[(j0 + l16) * 3 + 0], y = tc[(j0 + l16) * 3 + 1];
            tsq_part = x * x + y * y; b0 = x; b1 = y;
        } else {
            float z = tc[(j0 + l16) * 3 + 2];
            tsq_part = z * z; b0 = z; b1 = 0.0f;
        }
        float tsq_all = tsq_part + __shfl_xor(tsq_part, 16, 32); // tsq[col = l16]
        bool  mvalid  = mk[j0 + l16] != 0;

        v2f Bm; Bm.x = b0; Bm.y = b1;

        v8f C;
#pragma unroll
        for (int r = 0; r < 8; ++r) C[r] = p8[r] + tsq_all;

        // D = (-2*P) x T^T + (psq + tsq) = dist^2 (pre-clamp)
        v8f Dm = __builtin_amdgcn_wmma_f32_16x16x4_f32(
            /*neg_a=*/false, A, /*neg_b=*/false, Bm,
            /*c_mod=*/(short)0, C, /*reuse_a=*/false, /*reuse_b=*/false);

        float colmin = __builtin_inff();
#pragma unroll
        for (int r = 0; r < 8; ++r) {
            float d = Dm[r];
            d = d < 0.0f ? 0.0f : d;                 // clip(., 0, None)
            colmin = fminf(colmin, d);               // t2p uses UNmasked dist
            float dmask = mvalid ? d : __builtin_inff();
            if (dmask < rowmin[r]) { rowmin[r] = dmask; rowidx[r] = j0 + l16; }
        }
        // combine the two 8-row halves of this tile's column mins
        colmin = fminf(colmin, __shfl_xor(colmin, 16, 32));
        if (!half)
            atomicMin(&t2p[j0 + l16], __float_as_uint(colmin));
    }

    // ---- reduce row min/argmin across the 16 lanes of each half ----
#pragma unroll
    for (int m = 1; m < 16; m <<= 1) {
#pragma unroll
        for (int r = 0; r < 8; ++r) {
            float om = __shfl_xor(rowmin[r], m, 32);
            int   oi = __shfl_xor(rowidx[r], m, 32);
            if (om < rowmin[r] || (om == rowmin[r] && oi < rowidx[r])) {
                rowmin[r] = om; rowidx[r] = oi;      // first-occurrence argmin
            }
        }
    }
    if (l16 == 0) {
#pragma unroll
        for (int r = 0; r < 8; ++r) {
            int row = i0 + half * 8 + r;
            p2t[(size_t)b * K_ + row]  = rowmin[r];
            amin[(size_t)b * K_ + row] = rowidx[r];
        }
    }
}

// One block per batch: coord-loss partials + gathered smooth-L1 feature loss.
__global__ __launch_bounds__(256)
void chamfer_finalize_kernel(const float* __restrict__ p2t,
                             const int*   __restrict__ amin,
                             const unsigned int* __restrict__ t2p_bits,
                             const unsigned char* __restrict__ mask,
                             const float* __restrict__ pred_feat,  // (B,K,D)
                             const float* __restrict__ tgt_feat,   // (B,K,D)
                             float* __restrict__ accum)            // [coord, fnum, fden]
{
    __shared__ float s0[256], s1[256], s2[256], s3[256], s4[256];
    const int b = blockIdx.x, t = threadIdx.x;
    float sp = 0.f, st = 0.f, sc = 0.f, fn = 0.f, fd = 0.f;

    for (int i = t; i < K_; i += 256) {
        size_t o = (size_t)b * K_ + i;
        sp += p2t[o];
        float m = mask[o] ? 1.0f : 0.0f;
        st += __uint_as_float(t2p_bits[o]) * m;
        sc += m;

        int   j  = amin[o];
        float mv = mask[(size_t)b * K_ + j] ? 1.0f : 0.0f;
        const float* pf = pred_feat + o * D_;
        const float* tf = tgt_feat + ((size_t)b * K_ + j) * D_;
        float s = 0.f;
#pragma unroll
        for (int d = 0; d < D_; ++d) {
            float diff = pf[d] - tf[d];
            float ad = fabsf(diff);
            s += (ad < 1.0f) ? 0.5f * diff * diff : ad - 0.5f;
        }
        fn += s * (1.0f / (float)D_) * mv;
        fd += mv;
    }
    s0[t] = sp; s1[t] = st; s2[t] = sc; s3[t] = fn; s4[t] = fd;
    __syncthreads();
    for (int w = 128; w > 0; w >>= 1) {
        if (t < w) { s0[t]+=s0[t+w]; s1[t]+=s1[t+w]; s2[t]+=s2[t+w];
                     s3[t]+=s3[t+w]; s4[t]+=s4[t+w]; }
        __syncthreads();
    }
    if (t == 0) {
        float cnt = s2[0] < 1.0f ? 1.0f : s2[0];
        atomicAdd(&accum[0], s0[0] / (float)K_ + s1[0] / cnt); // per-batch coord loss
        atomicAdd(&accum[1], s3[0]);
        atomicAdd(&accum[2], s4[0]);
    }
}

__global__ void chamfer_writeout_kernel(const float* __restrict__ accum,
                                        float* __restrict__ out)
{
    if (threadIdx.x == 0 && blockIdx.x == 0) {
        float coord = accum[0] / (float)B_;
        float fden  = accum[2] < 1.0f ? 1.0f : accum[2];
        float feat  = accum[1] / fden;
        out[0] = 1.0f * (1.0f * coord + 0.1f * feat);  // total_loss
        out[1] = coord;                                 // coord_loss
        out[2] = feat;                                  // feat_loss
    }
}

extern "C" void kernel_launch(void* const* d_in, const int* in_sizes, int n_in,
                              void* d_out, int out_size, void* d_ws, size_t ws_size,
                              hipStream_t stream)
{
    const float* pred_coord = (const float*)d_in[0];
    const float* tgt_coord  = (const float*)d_in[1];
    const float* pred_feat  = (const float*)d_in[2];
    const float* tgt_feat   = (const float*)d_in[3];
    // jnp bool arrays are 1 byte/element; treat as raw bytes.
    const unsigned char* mask = (const unsigned char*)d_in[4];

    // Workspace layout (all 4-byte aligned):
    char* ws = (char*)d_ws;
    const size_t BK = (size_t)B_ * K_;
    float*        p2t      = (float*)ws;                         // BK floats
    int*          amin     = (int*)(ws + BK * 4);                // BK ints
    unsigned int* t2p_bits = (unsigned int*)(ws + BK * 8);       // BK uints
    float*        accum    = (float*)(ws + BK * 12);             // 4 floats

    // 1) init
    {
        int n = (int)BK;
        chamfer_init_kernel<<<(n + 255) / 256, 256, 0, stream>>>(t2p_bits, accum, n);
    }
    // 2) WMMA chamfer: B*K/16 = 4096 waves, 8 waves (256 thr) per block
    {
        int waves = (int)(BK / 16);
        chamfer_wmma_kernel<<<waves / 8, 256, 0, stream>>>(
            pred_coord, tgt_coord, mask, p2t, amin, t2p_bits);
    }
    // 3) per-batch finalize
    chamfer_finalize_kernel<<<B_, 256, 0, stream>>>(
        p2t, amin, t2p_bits, mask, pred_feat, tgt_feat, accum);
    // 4) write 3 scalars
    chamfer_writeout_kernel<<<1, 32, 0, stream>>>(accum, (float*)d_out);
}